// GCN_82291573391519
// MI455X (gfx1250) — compile-verified
//
#include <hip/hip_runtime.h>
#include <hip/hip_bf16.h>

typedef __attribute__((ext_vector_type(2))) float v2f;
typedef __attribute__((ext_vector_type(8))) float v8f;

// ---------------------------------------------------------------------------
// Degree / normalization:  deg[i] = 1 (self loop) + sum_{e: dst==i} ew[e]
// then dinv[i] = rsqrt(deg[i])   (deg >= 1 so always valid)
// ---------------------------------------------------------------------------
__global__ void k_deg_init(float* __restrict__ deg, int n) {
  int i = blockIdx.x * blockDim.x + threadIdx.x;
  if (i < n) deg[i] = 1.0f;
}

__global__ void k_deg_acc(const int* __restrict__ dst, const float* __restrict__ ew,
                          float* __restrict__ deg, int E) {
  int e = blockIdx.x * blockDim.x + threadIdx.x;
  if (e < E) atomicAdd(&deg[dst[e]], ew[e]);
}

__global__ void k_dinv(float* __restrict__ deg, int n) {
  int i = blockIdx.x * blockDim.x + threadIdx.x;
  if (i < n) deg[i] = rsqrtf(deg[i]);
}

// ---------------------------------------------------------------------------
// GEMM1: h1[N,64] = x[N,128] @ W1[128,64], fp32 WMMA 16x16x4.
// 256 threads = 8 waves; each wave owns one 16-row tile and 4 column tiles.
//
// W1 is staged raw into LDS with GLOBAL_LOAD_ASYNC_TO_LDS_B128 (ASYNCcnt),
// then pre-swizzled into per-lane B-fragment order so the hot loop does one
// ds_load_b64 per (k-step, n-tile) straight into an even VGPR pair.
//
// Fragment layouts (ISA 7.12.2, 32-bit 16x4):
//  A: lane<16 -> M=lane,   K={k,k+1};  lane>=16 -> M=lane-16, K={k+2,k+3}
//  B (mirrored): lane<16 -> N=lane, K={k,k+1}; lane>=16 -> N=lane-16, K={k+2,k+3}
//  C/D: VGPR r -> M = r + (lane>=16 ? 8 : 0), N = lane&15
// ---------------------------------------------------------------------------
__global__ void __launch_bounds__(256) k_gemm1(const float* __restrict__ x,
                                               const float* __restrict__ W1,
                                               float* __restrict__ h1, int ntiles) {
  __shared__ float sRaw[128 * 64];        // 32 KB raw W1
  __shared__ float sWs[32 * 4 * 64];      // 32 KB swizzled fragments [kstep][n][lane]{2}

  // ---- stage raw W1 via async global->LDS DMA (b128 = 4 floats/chunk) ----
  for (int i = threadIdx.x; i < 2048; i += 256) {
    unsigned     ldsoff = (unsigned)(size_t)&sRaw[i * 4];
    const float* gaddr  = W1 + (size_t)i * 4;
    asm volatile("global_load_async_to_lds_b128 %0, %1, off"
                 :: "v"(ldsoff), "v"(gaddr) : "memory");
  }
  asm volatile("s_wait_asynccnt 0x0" ::: "memory");
  __syncthreads();

  // ---- swizzle into B-fragment order: element i = (kstep, n, lane) ----
  for (int i = threadIdx.x; i < 4096; i += 256) {
    int kstep = i >> 7;
    int rem   = i & 127;
    int n     = rem >> 5;
    int ln    = rem & 31;
    int col   = ln & 15;
    int k0    = kstep * 4 + ((ln >= 16) ? 2 : 0);
    float2 v;
    v.x = sRaw[k0 * 64 + n * 16 + col];
    v.y = sRaw[(k0 + 1) * 64 + n * 16 + col];
    *(float2*)&sWs[(size_t)i * 2] = v;
  }
  __syncthreads();

  int wave = threadIdx.x >> 5;
  int lane = threadIdx.x & 31;
  int tile = blockIdx.x * 8 + wave;
  if (tile < ntiles) {
    int col  = lane & 15;
    int koff = (lane >= 16) ? 2 : 0;
    const float* arow  = x + (size_t)(tile * 16 + col) * 128;
    const float* bbase = &sWs[lane * 2];

    v8f acc0 = {}, acc1 = {}, acc2 = {}, acc3 = {};
    for (int kstep = 0; kstep < 32; ++kstep) {
      __builtin_prefetch(arow + kstep * 4 + 32, 0, 0);   // global_prefetch_b8
      v2f a;
      a.x = arow[kstep * 4 + koff];
      a.y = arow[kstep * 4 + koff + 1];
      const float* bk = bbase + kstep * 256;
      v2f b0 = *(const v2f*)(bk);
      v2f b1 = *(const v2f*)(bk + 64);
      v2f b2 = *(const v2f*)(bk + 128);
      v2f b3 = *(const v2f*)(bk + 192);
      acc0 = __builtin_amdgcn_wmma_f32_16x16x4_f32(false, a, false, b0, (short)0, acc0, false, false);
      acc1 = __builtin_amdgcn_wmma_f32_16x16x4_f32(false, a, false, b1, (short)0, acc1, false, false);
      acc2 = __builtin_amdgcn_wmma_f32_16x16x4_f32(false, a, false, b2, (short)0, acc2, false, false);
      acc3 = __builtin_amdgcn_wmma_f32_16x16x4_f32(false, a, false, b3, (short)0, acc3, false, false);
    }
    int rbase = tile * 16 + ((lane >= 16) ? 8 : 0);
    for (int r = 0; r < 8; ++r) {
      float* o = h1 + (size_t)(rbase + r) * 64 + col;
      o[0]  = acc0[r];
      o[16] = acc1[r];
      o[32] = acc2[r];
      o[48] = acc3[r];
    }
  }
}

// ---------------------------------------------------------------------------
// GEMM2: h2[N,16] = relu(agg1[N,64] + b1) @ W2[64,16], bias+relu fused into A.
// W2 (4 KB) swizzled into fragment order directly from global.
// ---------------------------------------------------------------------------
__global__ void __launch_bounds__(256) k_gemm2(const float* __restrict__ agg1,
                                               const float* __restrict__ bias1,
                                               const float* __restrict__ W2,
                                               float* __restrict__ h2, int ntiles) {
  __shared__ float sWs[16 * 64];   // [kstep(16)][lane(32)]{2} = 1024 floats
  __shared__ float sB[64];

  for (int i = threadIdx.x; i < 512; i += 256) {
    int kstep = i >> 5;
    int ln    = i & 31;
    int col   = ln & 15;
    int k0    = kstep * 4 + ((ln >= 16) ? 2 : 0);
    float2 v;
    v.x = W2[k0 * 16 + col];
    v.y = W2[(k0 + 1) * 16 + col];
    *(float2*)&sWs[(size_t)i * 2] = v;
  }
  if (threadIdx.x < 64) sB[threadIdx.x] = bias1[threadIdx.x];
  __syncthreads();

  int wave = threadIdx.x >> 5;
  int lane = threadIdx.x & 31;
  int tile = blockIdx.x * 8 + wave;
  if (tile < ntiles) {
    int col  = lane & 15;
    int koff = (lane >= 16) ? 2 : 0;
    const float* arow  = agg1 + (size_t)(tile * 16 + col) * 64;
    const float* bbase = &sWs[lane * 2];
    v8f acc = {};
    for (int kstep = 0; kstep < 16; ++kstep) {
      int k = kstep * 4 + koff;
      v2f a;
      a.x = fmaxf(arow[k]     + sB[k],     0.0f);
      a.y = fmaxf(arow[k + 1] + sB[k + 1], 0.0f);
      v2f b = *(const v2f*)(bbase + kstep * 64);
      acc = __builtin_amdgcn_wmma_f32_16x16x4_f32(false, a, false, b, (short)0, acc, false, false);
    }
    int rbase = tile * 16 + ((lane >= 16) ? 8 : 0);
    for (int r = 0; r < 8; ++r)
      h2[(size_t)(rbase + r) * 16 + col] = acc[r];
  }
}

// ---------------------------------------------------------------------------
// Aggregation: out = dinv^2 * h (self loop, also initializes buffer), then
// edge scatter out[dst] += (dinv[src]*ew*dinv[dst]) * h[src] via f32 atomics.
// All hot arrays (h, out, indices) are L2-resident (192 MB) on MI455X.
// ---------------------------------------------------------------------------
__global__ void k_selfinit(const float* __restrict__ h, const float* __restrict__ dinv,
                           float* __restrict__ out, int n, int C) {
  size_t i = (size_t)blockIdx.x * blockDim.x + threadIdx.x;
  if (i < (size_t)n * C) {
    int node = (int)(i / C);
    float di = dinv[node];
    out[i] = di * di * h[i];
  }
}

__global__ void k_scatter64(const int* __restrict__ src, const int* __restrict__ dst,
                            const float* __restrict__ ew, const float* __restrict__ dinv,
                            const float* __restrict__ h, float* __restrict__ out, int E) {
  size_t tid = (size_t)blockIdx.x * blockDim.x + threadIdx.x;
  int e    = (int)(tid >> 5);
  int lane = (int)(tid & 31);
  if (e >= E) return;
  int s = src[e], d = dst[e];
  float coef = dinv[s] * ew[e] * dinv[d];
  const float2 v = *(const float2*)(h + (size_t)s * 64 + lane * 2);
  float* o = out + (size_t)d * 64 + lane * 2;
  atomicAdd(o,     coef * v.x);
  atomicAdd(o + 1, coef * v.y);
}

__global__ void k_scatter16(const int* __restrict__ src, const int* __restrict__ dst,
                            const float* __restrict__ ew, const float* __restrict__ dinv,
                            const float* __restrict__ h, float* __restrict__ out, int E) {
  size_t tid = (size_t)blockIdx.x * blockDim.x + threadIdx.x;
  int e = (int)(tid >> 4);
  int c = (int)(tid & 15);
  if (e >= E) return;
  int s = src[e], d = dst[e];
  float coef = dinv[s] * ew[e] * dinv[d];
  atomicAdd(out + (size_t)d * 16 + c, coef * h[(size_t)s * 16 + c]);
}

// ---------------------------------------------------------------------------
// Final: row-wise  out = (row + b2) - max - log(sum(exp(...)))  (in place)
// ---------------------------------------------------------------------------
__global__ void k_logsoftmax16(float* __restrict__ out, const float* __restrict__ bias, int n) {
  int i = blockIdx.x * blockDim.x + threadIdx.x;
  if (i >= n) return;
  float* row = out + (size_t)i * 16;
  float v[16];
  float m = -3.0e38f;
  for (int c = 0; c < 16; ++c) { v[c] = row[c] + bias[c]; m = fmaxf(m, v[c]); }
  float ssum = 0.0f;
  for (int c = 0; c < 16; ++c) ssum += expf(v[c] - m);
  float ls = m + logf(ssum);
  for (int c = 0; c < 16; ++c) row[c] = v[c] - ls;
}

// ---------------------------------------------------------------------------
extern "C" void kernel_launch(void* const* d_in, const int* in_sizes, int n_in,
                              void* d_out, int out_size, void* d_ws, size_t ws_size,
                              hipStream_t stream) {
  const float* x    = (const float*)d_in[0];
  const int*   eidx = (const int*)d_in[1];   // [2, E]
  const float* ew   = (const float*)d_in[2];
  const float* W1   = (const float*)d_in[3];
  const float* b1   = (const float*)d_in[4];
  const float* W2   = (const float*)d_in[5];
  const float* b2   = (const float*)d_in[6];

  const int N = in_sizes[0] / 128;
  const int E = in_sizes[2];
  const int* src = eidx;
  const int* dst = eidx + E;

  float* ws   = (float*)d_ws;
  float* dinv = ws;                               // N
  float* h1   = dinv + (size_t)N;                 // N*64
  float* agg1 = h1   + (size_t)N * 64;            // N*64
  float* h2   = agg1 + (size_t)N * 64;            // N*16
  float* out  = (float*)d_out;                    // N*16

  const int T = 256;
  const int ntiles = (N + 15) / 16;               // N = 100000 -> exact multiple of 16
  const int gemm_blocks = (ntiles + 7) / 8;

  // 1) normalization coefficients
  k_deg_init<<<(N + T - 1) / T, T, 0, stream>>>(dinv, N);
  k_deg_acc<<<(E + T - 1) / T, T, 0, stream>>>(dst, ew, dinv, E);
  k_dinv<<<(N + T - 1) / T, T, 0, stream>>>(dinv, N);

  // 2) h1 = x @ W1   (fp32 WMMA, async W staging)
  k_gemm1<<<gemm_blocks, T, 0, stream>>>(x, W1, h1, ntiles);

  // 3) agg1 = D^-1/2 (A+I) D^-1/2 h1
  {
    size_t tot = (size_t)N * 64;
    k_selfinit<<<(unsigned)((tot + T - 1) / T), T, 0, stream>>>(h1, dinv, agg1, N, 64);
    size_t thr = (size_t)E * 32;
    k_scatter64<<<(unsigned)((thr + T - 1) / T), T, 0, stream>>>(src, dst, ew, dinv, h1, agg1, E);
  }

  // 4) h2 = relu(agg1 + b1) @ W2   (fp32 WMMA, fused bias+relu)
  k_gemm2<<<gemm_blocks, T, 0, stream>>>(agg1, b1, W2, h2, ntiles);

  // 5) out = D^-1/2 (A+I) D^-1/2 h2
  {
    size_t tot = (size_t)N * 16;
    k_selfinit<<<(unsigned)((tot + T - 1) / T), T, 0, stream>>>(h2, dinv, out, N, 16);
    size_t thr = (size_t)E * 16;
    k_scatter16<<<(unsigned)((thr + T - 1) / T), T, 0, stream>>>(src, dst, ew, dinv, h2, out, E);
  }

  // 6) bias + log_softmax (in place on d_out)
  k_logsoftmax16<<<(N + T - 1) / T, T, 0, stream>>>(out, b2, N);
}